// MatchLSTM_51307679318180
// MI455X (gfx1250) — compile-verified
//
#include <hip/hip_runtime.h>
#include <hip/hip_bf16.h>

// ---------------------------------------------------------------------------
// Match-LSTM for MI455X (gfx1250), wave32, fp32 WMMA (V_WMMA_F32_16X16X4_F32).
//
// Shapes: B=64, P=64 (65 premise steps), H=64, EMB=768, HID=512, 4H=2048, NC=3
// Strategy:
//   - pack all weight matrices once into k4-interleaved B-operand layout:
//       PB[((k>>2)*N + n)*4 + (k&3)] = W[n][k]
//     so each WMMA lane's B pair (k+kb, k+kb+1) is one 8B-aligned b64 load
//     (A pairs are already contiguous) -> 2x fewer B-load instructions in the
//     latency-bound recurrent inner loops.
//   - huge parallel GEMMs (x-projections, Ws*h_s) as one-wave-per-16x16-tile
//   - sequential LSTMs as single 1024-thread persistent workgroups; each wave
//     owns 4 (batch,hid) 16x16 tiles *and all 4 gate tiles* for them, so the
//     cell nonlinearity + c-state stay entirely in that wave's registers.
// ---------------------------------------------------------------------------

typedef float v2f __attribute__((ext_vector_type(2)));
typedef float v8f __attribute__((ext_vector_type(8)));

#define WMMA4(a, b, c) \
  __builtin_amdgcn_wmma_f32_16x16x4_f32(false, (a), false, (b), (short)0, (c), false, false)

__device__ __forceinline__ v8f vzero8() {
  v8f z = {0.f, 0.f, 0.f, 0.f, 0.f, 0.f, 0.f, 0.f};
  return z;
}

__device__ __forceinline__ float sigm(float x) { return 1.0f / (1.0f + __expf(-x)); }

// Accumulate one 16x16 f32 tile.
//   A  : row-major, pre-offset to (row m0, k 0), leading dim lda
//   PB : k4-packed B operand, pre-offset by column n0*4; ldbN = total N of PB
// A lane layout (ISA 7.12.2): lanes0-15 hold K=0,1; lanes16-31 hold K=2,3.
__device__ __forceinline__ v8f wmma_tile(const float* __restrict__ A, int lda,
                                         const float* __restrict__ PB, int ldbN,
                                         int K, v8f acc) {
  const int lane = threadIdx.x & 31;
  const int mr = lane & 15;            // A row within tile / B col within tile
  const int kb = (lane >> 4) << 1;     // k sub-offset (0 or 2)
  const float* arow = A + (size_t)mr * lda + kb;
  const float* bbase = PB + (size_t)mr * 4 + kb;
  const size_t bstep = (size_t)ldbN * 4;
  for (int k = 0; k < K; k += 4) {
    v2f a = *(const v2f*)(arow + k);
    v2f b = *(const v2f*)(bbase + (size_t)(k >> 2) * bstep);
    acc = WMMA4(a, b, acc);
  }
  return acc;
}

// ---------------------------------------------------------------------------
// Weight pack: W [N,K] row-major -> PB[((k>>2)*N + n)*4 + (k&3)] = W[n][k]
// ---------------------------------------------------------------------------
__global__ void pack_kernel(const float* __restrict__ src, float* __restrict__ dst,
                            int N, int K) {
  int idx = blockIdx.x * blockDim.x + threadIdx.x;
  if (idx >= N * K) return;
  int n = idx / K, k = idx - n * K;
  dst[((size_t)(k >> 2) * N + n) * 4 + (k & 3)] = src[idx];
}

// ---------------------------------------------------------------------------
// X-projection: Out[t*64+b, 0:2048] = pair[b, tok_off+t, :] @ Wih^T + (b1+b2)
// One wave per 16x16 output tile; K = 768; PB is k4-packed Wih (N=2048).
// ---------------------------------------------------------------------------
__global__ void xproj_kernel(const float* __restrict__ pair, const float* __restrict__ PB,
                             const float* __restrict__ b1, const float* __restrict__ b2,
                             float* __restrict__ Out, int T, int tok_off) {
  const int wg = blockIdx.x * (blockDim.x >> 5) + (threadIdx.x >> 5);
  const int tilesN = 2048 / 16;  // 128
  const int tiles = (T * 64 / 16) * tilesN;
  if (wg >= tiles) return;
  const int tr = wg / tilesN, tc = wg - tr * tilesN;
  const int m0 = tr * 16, n0 = tc * 16;
  const int lane = threadIdx.x & 31;
  const int mr = lane & 15;
  const int kb = (lane >> 4) << 1;
  const int row = m0 + mr;
  const int b = row & 63;            // batch
  const int t = row >> 6;            // timestep
  const float* Arow = pair + ((size_t)(b * 130 + tok_off + t)) * 768 + kb;
  const float* Bbase = PB + (size_t)(n0 + mr) * 4 + kb;
  v8f acc = vzero8();
  for (int k = 0; k < 768; k += 4) {
    v2f a = *(const v2f*)(Arow + k);
    v2f bb = *(const v2f*)(Bbase + (size_t)(k >> 2) * 8192);  // 2048*4
    acc = WMMA4(a, bb, acc);
  }
  const int nlo = lane & 15, mhi = (lane >> 4) << 3;
  const float bias = b1[n0 + nlo] + b2[n0 + nlo];
  for (int v = 0; v < 8; ++v)
    Out[(size_t)(m0 + mhi + v) * 2048 + n0 + nlo] = acc[v] + bias;
}

// ---------------------------------------------------------------------------
// Generic GEMM  D[M,N] = A[M,K] @ B^T (k4-packed, total width N)
// Used for ws_hs = h_s @ Ws^T.
// ---------------------------------------------------------------------------
__global__ void gemm_kernel(const float* __restrict__ A, int lda,
                            const float* __restrict__ PB, int ldbN,
                            float* __restrict__ D, int ldd, int M, int N, int K) {
  const int wg = blockIdx.x * (blockDim.x >> 5) + (threadIdx.x >> 5);
  const int tilesN = N >> 4;
  const int tiles = (M >> 4) * tilesN;
  if (wg >= tiles) return;
  const int tr = wg / tilesN, tc = wg - tr * tilesN;
  v8f acc = vzero8();
  acc = wmma_tile(A + (size_t)(tr * 16) * lda, lda, PB + (size_t)(tc * 16) * 4, ldbN, K, acc);
  const int lane = threadIdx.x & 31;
  const int nlo = lane & 15, mhi = (lane >> 4) << 3;
  float* Dp = D + (size_t)(tr * 16) * ldd + tc * 16;
  for (int v = 0; v < 8; ++v)
    Dp[(size_t)(mhi + v) * ldd + nlo] = acc[v];
}

// ---------------------------------------------------------------------------
// Sequential LSTM (premise: block 0, 65 steps; hypothesis: block 1, 64 steps).
// 1024 threads = 32 waves; wave w owns bh-tiles {w, w+32, w+64, w+96} of the
// 4x32 tile grid over [64 batch x 512 hid]; per tile it computes all 4 gate
// tiles (i,f,g,o) so the cell update is wave-local; c lives in registers.
// ---------------------------------------------------------------------------
__global__ void lstm_kernel(const float* __restrict__ Xp, const float* __restrict__ Xh,
                            const float* __restrict__ Phhp, const float* __restrict__ Phhh,
                            float* __restrict__ hs, float* __restrict__ ht) {
  const bool prem = (blockIdx.x == 0);
  const int T = prem ? 65 : 64;
  const float* X = prem ? Xp : Xh;
  const float* PB = prem ? Phhp : Phhh;  // k4-packed Whh^T, N=2048, K=512
  float* Hout = prem ? hs : ht;

  const int wid = threadIdx.x >> 5;
  const int lane = threadIdx.x & 31;
  const int mr = lane & 15, kb = (lane >> 4) << 1;
  const int nlo = lane & 15, mhi = (lane >> 4) << 3;

  float c[4][8];
  for (int i = 0; i < 4; ++i)
    for (int v = 0; v < 8; ++v) c[i][v] = 0.f;

  for (int t = 0; t < T; ++t) {
    const float* Hprev = Hout + (size_t)(t - 1) * 64 * 512;  // used only when t>0
    for (int i = 0; i < 4; ++i) {
      const int tI = wid + 32 * i;
      const int r = tI >> 5, cb = tI & 31;
      const int m0 = r * 16, n0h = cb * 16;
      v8f acc[4];
      for (int g = 0; g < 4; ++g) acc[g] = vzero8();
      if (t > 0) {
        const float* Arow = Hprev + (size_t)(m0 + mr) * 512 + kb;
        const float* B0 = PB + (size_t)(0 * 512 + n0h + mr) * 4 + kb;
        const float* B1 = PB + (size_t)(1 * 512 + n0h + mr) * 4 + kb;
        const float* B2 = PB + (size_t)(2 * 512 + n0h + mr) * 4 + kb;
        const float* B3 = PB + (size_t)(3 * 512 + n0h + mr) * 4 + kb;
        for (int k = 0; k < 512; k += 4) {
          const size_t bo = (size_t)(k >> 2) * 8192;  // 2048*4 per k-quad
          v2f a = *(const v2f*)(Arow + k);
          acc[0] = WMMA4(a, *(const v2f*)(B0 + bo), acc[0]);
          acc[1] = WMMA4(a, *(const v2f*)(B1 + bo), acc[1]);
          acc[2] = WMMA4(a, *(const v2f*)(B2 + bo), acc[2]);
          acc[3] = WMMA4(a, *(const v2f*)(B3 + bo), acc[3]);
        }
      }
      float gv[4][8];
      for (int g = 0; g < 4; ++g)
        for (int v = 0; v < 8; ++v)
          gv[g][v] = acc[g][v] +
                     X[(size_t)(t * 64 + m0 + mhi + v) * 2048 + g * 512 + n0h + nlo];
      for (int v = 0; v < 8; ++v) {
        float iv = sigm(gv[0][v]);
        float fv = sigm(gv[1][v]);
        float gg = tanhf(gv[2][v]);
        float ov = sigm(gv[3][v]);
        float cn = fv * c[i][v] + iv * gg;
        c[i][v] = cn;
        Hout[(size_t)(t * 64 + m0 + mhi + v) * 512 + n0h + nlo] = ov * tanhf(cn);
      }
    }
    __threadfence();
    __syncthreads();
  }
}

// ---------------------------------------------------------------------------
// Attention + match-LSTM loop (single 1024-thread workgroup, 64 steps).
// h_m double-buffered (hmA/hmB) so gates-GEMM reads never race state writes.
// ---------------------------------------------------------------------------
__global__ void match_kernel(const float* __restrict__ hs, const float* __restrict__ ht,
                             const float* __restrict__ wshs,
                             const float* __restrict__ PWt, const float* __restrict__ PWm,
                             const float* __restrict__ Pihm, const float* __restrict__ Phhm,
                             const float* __restrict__ w_e,
                             const float* __restrict__ bih, const float* __restrict__ bhh,
                             const float* __restrict__ fcw, const float* __restrict__ fcb,
                             float* __restrict__ proj, float* __restrict__ ebuf,
                             float* __restrict__ alpha, float* __restrict__ mk,
                             float* __restrict__ hmA, float* __restrict__ hmB,
                             float* __restrict__ out) {
  const int tid = threadIdx.x;
  const int wid = tid >> 5, lane = tid & 31;
  const int mr = lane & 15, kb = (lane >> 4) << 1;
  const int nlo = lane & 15, mhi = (lane >> 4) << 3;

  float cm[4][8];
  for (int i = 0; i < 4; ++i)
    for (int v = 0; v < 8; ++v) cm[i][v] = 0.f;
  for (int o = tid; o < 64 * 512; o += 1024) hmA[o] = 0.f;
  __threadfence();
  __syncthreads();

  for (int k = 0; k < 64; ++k) {
    const float* hprev = (k & 1) ? hmB : hmA;
    float* hnext = (k & 1) ? hmA : hmB;
    const float* htk = ht + (size_t)k * 64 * 512;

    // (a) proj[b,h] = h_t_k @ Wt^T + h_m @ Wm^T
    for (int i = 0; i < 4; ++i) {
      const int tI = wid + 32 * i;
      const int r = tI >> 5, cb = tI & 31;
      const int m0 = r * 16, n0 = cb * 16;
      v8f acc = vzero8();
      acc = wmma_tile(htk + (size_t)m0 * 512, 512, PWt + (size_t)n0 * 4, 512, 512, acc);
      acc = wmma_tile(hprev + (size_t)m0 * 512, 512, PWm + (size_t)n0 * 4, 512, 512, acc);
      for (int v = 0; v < 8; ++v)
        proj[(size_t)(m0 + mhi + v) * 512 + n0 + nlo] = acc[v];
    }
    __threadfence();
    __syncthreads();

    // (b) e[p,b] = w_e . tanh(ws_hs[p,b,:] + proj[b,:])   (wave per (p,b) pair)
    for (int idx = wid; idx < 65 * 64; idx += 32) {
      const int b = idx & 63;
      const float* wrow = wshs + (size_t)idx * 512;
      const float* prow = proj + (size_t)b * 512;
      float s = 0.f;
      for (int h = lane; h < 512; h += 32) s += w_e[h] * tanhf(wrow[h] + prow[h]);
      for (int off = 16; off > 0; off >>= 1) s += __shfl_xor(s, off, 32);
      if (lane == 0) ebuf[idx] = s;
    }
    __threadfence();
    __syncthreads();

    // (c) softmax over p (65) per batch column
    if (tid < 64) {
      float mx = -1e30f;
      for (int p = 0; p < 65; ++p) mx = fmaxf(mx, ebuf[p * 64 + tid]);
      float sum = 0.f;
      for (int p = 0; p < 65; ++p) {
        float ex = __expf(ebuf[p * 64 + tid] - mx);
        alpha[p * 64 + tid] = ex;
        sum += ex;
      }
      float inv = 1.f / sum;
      for (int p = 0; p < 65; ++p) alpha[p * 64 + tid] *= inv;
    }
    __threadfence();
    __syncthreads();

    // (d) a_k = sum_p alpha[p,b] h_s[p,b,:]; m_k = [a_k, h_t_k]
    for (int o = tid; o < 64 * 512; o += 1024) {
      const int b = o >> 9, h = o & 511;
      float s = 0.f;
      for (int p = 0; p < 65; ++p)
        s += alpha[p * 64 + b] * hs[((size_t)p * 64 + b) * 512 + h];
      mk[(size_t)b * 1024 + h] = s;
      mk[(size_t)b * 1024 + 512 + h] = htk[(size_t)b * 512 + h];
    }
    __threadfence();
    __syncthreads();

    // (e) match-LSTM cell: gates = m_k @ Wih_m^T + h_m @ Whh_m^T + bias
    for (int i = 0; i < 4; ++i) {
      const int tI = wid + 32 * i;
      const int r = tI >> 5, cb = tI & 31;
      const int m0 = r * 16, n0h = cb * 16;
      v8f acc[4];
      for (int g = 0; g < 4; ++g) acc[g] = vzero8();
      {
        const float* Arow = mk + (size_t)(m0 + mr) * 1024 + kb;
        const float* B0 = Pihm + (size_t)(0 * 512 + n0h + mr) * 4 + kb;
        const float* B1 = Pihm + (size_t)(1 * 512 + n0h + mr) * 4 + kb;
        const float* B2 = Pihm + (size_t)(2 * 512 + n0h + mr) * 4 + kb;
        const float* B3 = Pihm + (size_t)(3 * 512 + n0h + mr) * 4 + kb;
        for (int kk = 0; kk < 1024; kk += 4) {
          const size_t bo = (size_t)(kk >> 2) * 8192;
          v2f a = *(const v2f*)(Arow + kk);
          acc[0] = WMMA4(a, *(const v2f*)(B0 + bo), acc[0]);
          acc[1] = WMMA4(a, *(const v2f*)(B1 + bo), acc[1]);
          acc[2] = WMMA4(a, *(const v2f*)(B2 + bo), acc[2]);
          acc[3] = WMMA4(a, *(const v2f*)(B3 + bo), acc[3]);
        }
      }
      {
        const float* Arow = hprev + (size_t)(m0 + mr) * 512 + kb;
        const float* B0 = Phhm + (size_t)(0 * 512 + n0h + mr) * 4 + kb;
        const float* B1 = Phhm + (size_t)(1 * 512 + n0h + mr) * 4 + kb;
        const float* B2 = Phhm + (size_t)(2 * 512 + n0h + mr) * 4 + kb;
        const float* B3 = Phhm + (size_t)(3 * 512 + n0h + mr) * 4 + kb;
        for (int kk = 0; kk < 512; kk += 4) {
          const size_t bo = (size_t)(kk >> 2) * 8192;
          v2f a = *(const v2f*)(Arow + kk);
          acc[0] = WMMA4(a, *(const v2f*)(B0 + bo), acc[0]);
          acc[1] = WMMA4(a, *(const v2f*)(B1 + bo), acc[1]);
          acc[2] = WMMA4(a, *(const v2f*)(B2 + bo), acc[2]);
          acc[3] = WMMA4(a, *(const v2f*)(B3 + bo), acc[3]);
        }
      }
      float gv[4][8];
      for (int g = 0; g < 4; ++g) {
        const float bias = bih[g * 512 + n0h + nlo] + bhh[g * 512 + n0h + nlo];
        for (int v = 0; v < 8; ++v) gv[g][v] = acc[g][v] + bias;
      }
      for (int v = 0; v < 8; ++v) {
        float iv = sigm(gv[0][v]);
        float fv = sigm(gv[1][v]);
        float gg = tanhf(gv[2][v]);
        float ov = sigm(gv[3][v]);
        float cn = fv * cm[i][v] + iv * gg;
        cm[i][v] = cn;
        hnext[(size_t)(m0 + mhi + v) * 512 + n0h + nlo] = ov * tanhf(cn);
      }
    }
    __threadfence();
    __syncthreads();
  }

  // final classifier: out = h_last @ fc_w^T + fc_b   (h_last == hmA after 64 steps)
  if (tid < 64 * 3) {
    const int b = tid / 3, cc = tid - b * 3;
    float s = fcb[cc];
    const float* hrow = hmA + (size_t)b * 512;
    const float* wrow = fcw + (size_t)cc * 512;
    for (int h = 0; h < 512; ++h) s += hrow[h] * wrow[h];
    out[b * 3 + cc] = s;
  }
}

// ---------------------------------------------------------------------------
extern "C" void kernel_launch(void* const* d_in, const int* in_sizes, int n_in,
                              void* d_out, int out_size, void* d_ws, size_t ws_size,
                              hipStream_t stream) {
  (void)in_sizes; (void)n_in; (void)out_size; (void)ws_size;
  const float* pair  = (const float*)d_in[0];
  // d_in[1], d_in[2]: premise_len / hypothesis_len (hardcoded 64/64)
  const float* w_e   = (const float*)d_in[3];
  const float* Ws    = (const float*)d_in[4];
  const float* Wt    = (const float*)d_in[5];
  const float* Wm    = (const float*)d_in[6];
  const float* fc_w  = (const float*)d_in[7];
  const float* fc_b  = (const float*)d_in[8];
  const float* Wih_p = (const float*)d_in[9];
  const float* Whh_p = (const float*)d_in[10];
  const float* bih_p = (const float*)d_in[11];
  const float* bhh_p = (const float*)d_in[12];
  const float* Wih_h = (const float*)d_in[13];
  const float* Whh_h = (const float*)d_in[14];
  const float* bih_h = (const float*)d_in[15];
  const float* bhh_h = (const float*)d_in[16];
  const float* Wih_m = (const float*)d_in[17];
  const float* Whh_m = (const float*)d_in[18];
  const float* bih_m = (const float*)d_in[19];
  const float* bhh_m = (const float*)d_in[20];
  float* out = (float*)d_out;

  // scratch layout (floats)
  float* w = (float*)d_ws;
  size_t o = 0;
  float* Pihp = w + o; o += (size_t)768 * 2048;
  float* Pihh = w + o; o += (size_t)768 * 2048;
  float* Phhp = w + o; o += (size_t)512 * 2048;
  float* Phhh = w + o; o += (size_t)512 * 2048;
  float* Pihm = w + o; o += (size_t)1024 * 2048;
  float* Phhm = w + o; o += (size_t)512 * 2048;
  float* PWs  = w + o; o += (size_t)512 * 512;
  float* PWtb = w + o; o += (size_t)512 * 512;
  float* PWmb = w + o; o += (size_t)512 * 512;
  float* Xp   = w + o; o += (size_t)65 * 64 * 2048;
  float* Xh   = w + o; o += (size_t)64 * 64 * 2048;
  float* hsb  = w + o; o += (size_t)65 * 64 * 512;
  float* htb  = w + o; o += (size_t)64 * 64 * 512;
  float* wshs = w + o; o += (size_t)65 * 64 * 512;
  float* proj = w + o; o += (size_t)64 * 512;
  float* ebuf = w + o; o += (size_t)65 * 64;
  float* alf  = w + o; o += (size_t)65 * 64;
  float* mk   = w + o; o += (size_t)64 * 1024;
  float* hmA  = w + o; o += (size_t)64 * 512;
  float* hmB  = w + o; o += (size_t)64 * 512;

  auto pk = [&](const float* src, float* dst, int N, int K) {
    int n = N * K;
    hipLaunchKernelGGL(pack_kernel, dim3((n + 255) / 256), dim3(256), 0, stream,
                       src, dst, N, K);
  };
  pk(Wih_p, Pihp, 2048, 768);
  pk(Wih_h, Pihh, 2048, 768);
  pk(Whh_p, Phhp, 2048, 512);
  pk(Whh_h, Phhh, 2048, 512);
  pk(Wih_m, Pihm, 2048, 1024);
  pk(Whh_m, Phhm, 2048, 512);
  pk(Ws, PWs, 512, 512);
  pk(Wt, PWtb, 512, 512);
  pk(Wm, PWmb, 512, 512);

  // x-projections (parallel, WMMA): premise tokens 1..65, hypothesis 66..129
  hipLaunchKernelGGL(xproj_kernel, dim3(4160), dim3(256), 0, stream,
                     pair, Pihp, bih_p, bhh_p, Xp, 65, 1);
  hipLaunchKernelGGL(xproj_kernel, dim3(4096), dim3(256), 0, stream,
                     pair, Pihh, bih_h, bhh_h, Xh, 64, 66);

  // sequential LSTMs (premise + hypothesis, one persistent workgroup each)
  hipLaunchKernelGGL(lstm_kernel, dim3(2), dim3(1024), 0, stream,
                     Xp, Xh, Phhp, Phhh, hsb, htb);

  // ws_hs = h_s @ Ws^T  : M=4160, N=512, K=512  -> 8320 tiles / 8 waves
  hipLaunchKernelGGL(gemm_kernel, dim3(1040), dim3(256), 0, stream,
                     hsb, 512, PWs, 512, wshs, 512, 4160, 512, 512);

  // attention + match-LSTM loop + classifier
  hipLaunchKernelGGL(match_kernel, dim3(1), dim3(1024), 0, stream,
                     hsb, htb, wshs, PWtb, PWmb, Pihm, Phhm,
                     w_e, bih_m, bhh_m, fc_w, fc_b,
                     proj, ebuf, alf, mk, hmA, hmB, out);
}